// SheafGluingValidator_8924942041560
// MI455X (gfx1250) — compile-verified
//
#include <hip/hip_runtime.h>
#include <hip/hip_bf16.h>
#include <math.h>

#define N_NODES 24
#define D 128
#define R 128
#define E_EDGES 552          // n*(n-1)
#define T_TRIS  12144        // 24*23*22
#define K_CNT   22           // k values per (i,j) pair
#define THRESHOLD 0.5f

// padded LDS row strides (floats) -> conflict-free WMMA operand fetches
#define SR_STR 132           // A operand rows: lane banks 4*lr (+2 K-half) distinct
#define SG_STR 144           // B operand rows: half-groups offset by 32 banks

typedef __attribute__((ext_vector_type(2))) float v2f;
typedef __attribute__((ext_vector_type(8))) float v8f;

__device__ __forceinline__ int edge_index(int a, int b) {
    int lo = a < b ? a : b;
    int hi = a < b ? b : a;
    int p = lo * (2 * N_NODES - lo - 1) / 2 + (hi - lo - 1);
    return 2 * p + (a > b ? 1 : 0);
}

__device__ __forceinline__ int kth_other(int kidx, int lo, int hi) {
    int k = kidx;
    if (k >= lo) ++k;
    if (k >= hi) ++k;
    return k;
}

// CDNA5 async global->LDS copy (ASYNCcnt-tracked), 16B per lane
__device__ __forceinline__ void async_b128(unsigned lds_byte_off, const float* gaddr) {
    asm volatile("global_load_async_to_lds_b128 %0, %1, off"
                 :: "v"(lds_byte_off), "v"(gaddr)
                 : "memory");
}
__device__ __forceinline__ void wait_async0() {
    asm volatile("s_wait_asynccnt 0x0" ::: "memory");
}

// stage a 128x128 f32 matrix (row-major, row stride 128) into LDS with padded
// row stride `lstr` (floats). 4096 16B chunks over 256 threads.
__device__ __forceinline__ void stage_async(unsigned lds_base_off,
                                            const float* src, int lstr, int tid) {
    #pragma unroll
    for (int it = 0; it < 16; ++it) {
        int ch = tid + it * 256;
        int r = ch >> 5;
        int c = (ch & 31) << 2;
        async_b128(lds_base_off + (unsigned)(r * lstr + c) * 4u, src + r * D + c);
    }
}

// ---------------------------------------------------------------------------
// Kernel 1: y[e] = R_e @ patches[src[e]]   (552 blocks x 128 threads)
// Exploits restricted_dst[e] == restricted_src[e^1].
// ---------------------------------------------------------------------------
__global__ __launch_bounds__(128) void matvec_kernel(
    const float* __restrict__ maps, const float* __restrict__ patches,
    float* __restrict__ y) {
    __shared__ float sp[D];
    const int e = blockIdx.x;
    int p = e >> 1;
    int i = 0, rem = p;
    while (rem >= (N_NODES - 1 - i)) { rem -= (N_NODES - 1 - i); ++i; }
    int j = i + 1 + rem;
    int s = (e & 1) ? j : i;

    const int tid = threadIdx.x;
    sp[tid] = patches[s * D + tid];
    __syncthreads();

    const float* row = maps + ((size_t)e * R + tid) * D;
    float acc = 0.f;
    #pragma unroll 8
    for (int d0 = 0; d0 < D; d0 += 4) {
        float4 m4 = *(const float4*)(row + d0);
        acc += m4.x * sp[d0] + m4.y * sp[d0 + 1] + m4.z * sp[d0 + 2] + m4.w * sp[d0 + 3];
    }
    y[(size_t)e * R + tid] = acc;
}

// ---------------------------------------------------------------------------
// Kernel 2: G[e] = R_e^T R_e via V_WMMA_F32_16X16X4_F32
// 552 blocks x 256 threads (8 waves). R_e staged async into padded LDS.
// A fragments (64 VGPRs) preloaded once per wave, reused across all 8
// column blocks; only B streams from LDS inside the MAC chain.
// ---------------------------------------------------------------------------
__global__ __launch_bounds__(256) void gram_kernel(
    const float* __restrict__ maps, float* __restrict__ G) {
    extern __shared__ float lds[];   // D * SG_STR floats
    const int e = blockIdx.x;
    const int tid = threadIdx.x;

    stage_async(0u, maps + (size_t)e * R * D, SG_STR, tid);
    wait_async0();
    __syncthreads();

    const int wave = tid >> 5;
    const int lane = tid & 31;
    const int half = lane >> 4;
    const int lr   = lane & 15;
    const int a0 = wave * 16;
    float* Ge = G + (size_t)e * D * D;

    // A[m][k] = M[k][a0+m] : constant per wave -> keep in registers
    v2f areg[32];
    #pragma unroll
    for (int q = 0; q < 32; ++q) {
        areg[q].x = lds[(4 * q + 2 * half    ) * SG_STR + a0 + lr];
        areg[q].y = lds[(4 * q + 2 * half + 1) * SG_STR + a0 + lr];
    }

    for (int bt = 0; bt < 8; ++bt) {
        const int b0 = bt * 16;
        v8f c = {};
        #pragma unroll
        for (int q = 0; q < 32; ++q) {
            v2f b;
            b.x = lds[(4 * q + 2 * half    ) * SG_STR + b0 + lr];
            b.y = lds[(4 * q + 2 * half + 1) * SG_STR + b0 + lr];
            c = __builtin_amdgcn_wmma_f32_16x16x4_f32(
                    false, areg[q], false, b, (short)0, c, false, false);
        }
        #pragma unroll
        for (int v = 0; v < 8; ++v) {
            int row = a0 + v + 8 * half;
            Ge[row * D + b0 + lr] = c[v];
        }
    }
}

// ---------------------------------------------------------------------------
// Kernel 3: one WG per ordered pair (i,j); loop over 22 k's.
//   composed = R_jk @ G_ij ;  accumulate sqrt(||composed - R_ik||_F^2)
// G_ij staged once (22x reuse); R_jk double-buffered via async-to-LDS so the
// next copy overlaps the current 2048-WMMA sweep. ~210KB LDS -> 1 WG/WGP.
// A fragments preloaded per k (row-contiguous v2f = ds_load_b64), B streams.
// ---------------------------------------------------------------------------
__global__ __launch_bounds__(256) void triangle_kernel(
    const float* __restrict__ maps, const float* __restrict__ G,
    float* __restrict__ comp_pair) {
    extern __shared__ float smem[];
    float* sG  = smem;                       // D * SG_STR
    float* sR0 = sG  + D * SG_STR;           // R * SR_STR
    float* sR1 = sR0 + (size_t)R * SR_STR;   // R * SR_STR
    __shared__ float red[256];

    const int p = blockIdx.x;                // ordered pair (i, j != i)
    const int i = p / (N_NODES - 1);
    const int jr = p % (N_NODES - 1);
    const int j = jr + (jr >= i ? 1 : 0);
    const int lo = i < j ? i : j;
    const int hi = i < j ? j : i;
    const int ij = edge_index(i, j);

    const int tid = threadIdx.x;
    const unsigned offG  = 0u;
    const unsigned offR0 = (unsigned)(D * SG_STR) * 4u;
    const unsigned offR1 = offR0 + (unsigned)(R * SR_STR) * 4u;

    // stage G_ij and first R_jk
    stage_async(offG, G + (size_t)ij * D * D, SG_STR, tid);
    {
        int k0v = kth_other(0, lo, hi);
        stage_async(offR0, maps + (size_t)edge_index(j, k0v) * R * D, SR_STR, tid);
    }
    wait_async0();
    __syncthreads();

    const int wave = tid >> 5;
    const int lane = tid & 31;
    const int half = lane >> 4;
    const int lr   = lane & 15;
    const int m0 = wave * 16;

    float acc = 0.f;   // thread 0 accumulates sum of per-triangle sqrt's

    #pragma unroll 1
    for (int kidx = 0; kidx < K_CNT; ++kidx) {
        const float* sR = (kidx & 1) ? sR1 : sR0;

        // prefetch next R_jk into the other buffer (overlaps compute below)
        if (kidx + 1 < K_CNT) {
            int kn = kth_other(kidx + 1, lo, hi);
            unsigned dstoff = (kidx & 1) ? offR0 : offR1;
            stage_async(dstoff, maps + (size_t)edge_index(j, kn) * R * D,
                        SR_STR, tid);
        }

        const int k = kth_other(kidx, lo, hi);
        const float* Rik = maps + (size_t)edge_index(i, k) * R * D;

        // A[m][kk] = R_jk[m0+m][kk] : constant per wave across all 8 column
        // blocks of this k -> preload to registers (8B-aligned v2f loads)
        v2f areg[32];
        const float* arow = sR + (m0 + lr) * SR_STR + 2 * half;
        #pragma unroll
        for (int q = 0; q < 32; ++q)
            areg[q] = *(const v2f*)(arow + 4 * q);

        float sq = 0.f;
        for (int bt = 0; bt < 8; ++bt) {
            const int n0 = bt * 16;
            v8f c = {};
            #pragma unroll
            for (int q = 0; q < 32; ++q) {
                // B[kk][n] = G_ij[kk][n0+n]
                v2f b;
                b.x = sG[(4 * q + 2 * half    ) * SG_STR + n0 + lr];
                b.y = sG[(4 * q + 2 * half + 1) * SG_STR + n0 + lr];
                c = __builtin_amdgcn_wmma_f32_16x16x4_f32(
                        false, areg[q], false, b, (short)0, c, false, false);
            }
            #pragma unroll
            for (int v = 0; v < 8; ++v) {
                int row = m0 + v + 8 * half;
                float delta = c[v] - Rik[row * D + n0 + lr];
                sq += delta * delta;
            }
        }

        // deterministic block reduction for this triangle
        red[tid] = sq;
        __syncthreads();
        for (int s = 128; s > 0; s >>= 1) {
            if (tid < s) red[tid] += red[tid + s];
            __syncthreads();
        }
        if (tid == 0) acc += sqrtf(red[0]);

        // prefetched buffer must be complete & visible before next iteration
        wait_async0();
        __syncthreads();
    }

    if (tid == 0) comp_pair[p] = acc;
}

// ---------------------------------------------------------------------------
// Kernel 4: defects + scalars + global_section (1 block, 256 threads)
// Output: [0..551] defects, [552] max, [553] mean, [554] score,
//         [555..682] global_section, [683] composition_defect
// ---------------------------------------------------------------------------
__global__ __launch_bounds__(256) void finalize_kernel(
    const float* __restrict__ y, const float* __restrict__ comp_pair,
    const float* __restrict__ patches, const float* __restrict__ W,
    float* __restrict__ out) {
    __shared__ float red[256];
    __shared__ float pbar[D];
    const int tid = threadIdx.x;

    float lmax = 0.f, lsum = 0.f;
    for (int e = tid; e < E_EDGES; e += 256) {
        const float* ya = y + (size_t)e * R;
        const float* yb = y + (size_t)(e ^ 1) * R;
        float s = 0.f;
        for (int r = 0; r < R; ++r) { float d = ya[r] - yb[r]; s += d * d; }
        float def = sqrtf(s);
        out[e] = def;
        lmax = fmaxf(lmax, def);
        lsum += def;
    }
    red[tid] = lmax; __syncthreads();
    for (int s = 128; s > 0; s >>= 1) {
        if (tid < s) red[tid] = fmaxf(red[tid], red[tid + s]);
        __syncthreads();
    }
    float max_def = red[0]; __syncthreads();

    red[tid] = lsum; __syncthreads();
    for (int s = 128; s > 0; s >>= 1) {
        if (tid < s) red[tid] += red[tid + s];
        __syncthreads();
    }
    float mean_def = red[0] / (float)E_EDGES; __syncthreads();

    if (tid == 0) {
        out[E_EDGES]     = max_def;
        out[E_EDGES + 1] = mean_def;
        out[E_EDGES + 2] = expf(-mean_def / THRESHOLD);
    }

    float csum = 0.f;
    for (int t = tid; t < E_EDGES; t += 256) csum += comp_pair[t];
    red[tid] = csum; __syncthreads();
    for (int s = 128; s > 0; s >>= 1) {
        if (tid < s) red[tid] += red[tid + s];
        __syncthreads();
    }
    if (tid == 0) out[683] = red[0] / (float)T_TRIS;
    __syncthreads();

    // global_section = (patches @ W^T).mean(axis=0) = W @ mean(patches)
    if (tid < D) {
        float s = 0.f;
        for (int n = 0; n < N_NODES; ++n) s += patches[n * D + tid];
        pbar[tid] = s / (float)N_NODES;
    }
    __syncthreads();
    if (tid < D) {
        const float* wrow = W + tid * D;
        float s = 0.f;
        for (int d = 0; d < D; ++d) s += wrow[d] * pbar[d];
        out[555 + tid] = s;
    }
}

extern "C" void kernel_launch(void* const* d_in, const int* in_sizes, int n_in,
                              void* d_out, int out_size, void* d_ws, size_t ws_size,
                              hipStream_t stream) {
    (void)in_sizes; (void)n_in; (void)out_size; (void)ws_size;
    const float* patches = (const float*)d_in[0];   // (24, 128)
    const float* maps    = (const float*)d_in[1];   // (552, 128, 128)
    const float* W       = (const float*)d_in[2];   // (128, 128)
    float* out = (float*)d_out;                     // 684 floats

    // workspace layout (floats): y[E*R] | G[E*D*D] | comp_pair[E]
    float* ws   = (float*)d_ws;
    float* y    = ws;                                // 70,656 floats
    float* G    = y + (size_t)E_EDGES * R;           // 9,043,968 floats
    float* comp = G + (size_t)E_EDGES * D * D;       // 552 floats

    const size_t gram_lds = (size_t)D * SG_STR * sizeof(float);                     // 73,728 B
    const size_t tri_lds  = ((size_t)D * SG_STR + 2u * R * SR_STR) * sizeof(float); // 208,896 B

    matvec_kernel  <<<E_EDGES, 128, 0, stream>>>(maps, patches, y);
    gram_kernel    <<<E_EDGES, 256, gram_lds, stream>>>(maps, G);
    triangle_kernel<<<E_EDGES, 256, tri_lds, stream>>>(maps, G, comp);
    finalize_kernel<<<1, 256, 0, stream>>>(y, comp, patches, W, out);
}